// travelKawaharaPinn_13202729468515
// MI455X (gfx1250) — compile-verified
//
#include <hip/hip_runtime.h>
#include <math.h>

// ---------------------------------------------------------------------------
// Travelling-wave Kawahara PINN: degree-5 Taylor-jet forward pass through a
// 2->64->64->64->64->1 tanh MLP, matmuls on V_WMMA_F32_16X16X4_F32 (full f32
// precision -- 5th derivatives cannot survive f16). One wave = 16 points.
// Layer loop is peeled so weight loads resolve to addrspace(1) -> global_load
// (keeps B-operand traffic off the FLAT/LDS pipe; A stays on ds_load_b64).
// ---------------------------------------------------------------------------

typedef __attribute__((ext_vector_type(2))) float v2f;
typedef __attribute__((ext_vector_type(8))) float v8f;

#define RS 68          // padded row stride (floats) for 64-wide activation rows
#define NW 2           // waves per block
#define NEQ 32768
#define NIB 2048

__device__ __forceinline__ void tanh_jet5(const float a[6], float y[6]) {
    // Taylor-coefficient propagation through tanh via y' = (1 - y^2) a'
    float y0 = tanhf(a[0]);
    float w0 = 1.0f - y0 * y0;
    float y1 = w0 * a[1];
    float w1 = -2.0f * y0 * y1;
    float y2 = w0 * a[2] + 0.5f * w1 * a[1];
    float w2 = -(2.0f * y0 * y2 + y1 * y1);
    float y3 = w0 * a[3] + (2.0f / 3.0f) * w1 * a[2] + (1.0f / 3.0f) * w2 * a[1];
    float w3 = -2.0f * (y0 * y3 + y1 * y2);
    float y4 = w0 * a[4] + 0.75f * w1 * a[3] + 0.5f * w2 * a[2] + 0.25f * w3 * a[1];
    float w4 = -(2.0f * y0 * y4 + 2.0f * y1 * y3 + y2 * y2);
    float y5 = w0 * a[5] + 0.8f * w1 * a[4] + 0.6f * w2 * a[3]
             + 0.4f * w3 * a[2] + 0.2f * w4 * a[1];
    y[0] = y0; y[1] = y1; y[2] = y2; y[3] = y3; y[4] = y4; y[5] = y5;
}

// One [64->64] layer with tanh-jet, 6 Taylor coefficient planes, on WMMA f32.
// Called with weight pointers straight from kernel args so clang infers
// addrspace(1) and emits global_load for the B operand.
__device__ __forceinline__ void layer64_jet(
    const float* __restrict__ Wg, const float* __restrict__ bg,
    float (&S)[6][16 * RS], int m16, int hi)
{
    v8f acc[6][4];
    #pragma unroll
    for (int nt = 0; nt < 4; ++nt) {
        float bv = bg[nt * 16 + m16];         // bias keyed by N = lane%16
        #pragma unroll
        for (int r = 0; r < 8; ++r) acc[0][nt][r] = bv;
        #pragma unroll
        for (int c = 1; c < 6; ++c)
            #pragma unroll
            for (int r = 0; r < 8; ++r) acc[c][nt][r] = 0.f;
    }

    #pragma unroll 1
    for (int kt = 0; kt < 16; ++kt) {
        const int k0 = kt * 4 + hi * 2;       // A layout: lane half picks K pair
        v2f A[6];
        #pragma unroll
        for (int c = 0; c < 6; ++c)
            A[c] = *reinterpret_cast<const v2f*>(&S[c][m16 * RS + k0]);
        #pragma unroll
        for (int nt = 0; nt < 4; ++nt) {
            const int nb = nt * 16 + m16;
            v2f B;
            B.x = Wg[k0 * 64 + nb];           // B layout mirrors A: v0<->k0, v1<->k0+1
            B.y = Wg[(k0 + 1) * 64 + nb];
            #pragma unroll
            for (int c = 0; c < 6; ++c)
                acc[c][nt] = __builtin_amdgcn_wmma_f32_16x16x4_f32(
                    false, A[c], false, B, (short)0, acc[c][nt], false, false);
        }
    }

    // tanh-jet on accumulators (per-lane: all 6 coefficients co-resident),
    // scatter back to LDS in activation layout (= transpose for next layer)
    #pragma unroll
    for (int nt = 0; nt < 4; ++nt) {
        #pragma unroll
        for (int r = 0; r < 8; ++r) {
            float a[6], y[6];
            #pragma unroll
            for (int c = 0; c < 6; ++c) a[c] = acc[c][nt][r];
            tanh_jet5(a, y);
            const int m = r + 8 * hi;         // D layout row
            const int n = nt * 16 + m16;      // D layout col
            #pragma unroll
            for (int c = 0; c < 6; ++c) S[c][m * RS + n] = y[c];
        }
    }
}

// Forward-only variant (coefficient-0 plane).
__device__ __forceinline__ void layer64_fwd(
    const float* __restrict__ Wg, const float* __restrict__ bg,
    float (&S)[16 * RS], int m16, int hi)
{
    v8f acc[4];
    #pragma unroll
    for (int nt = 0; nt < 4; ++nt) {
        float bv = bg[nt * 16 + m16];
        #pragma unroll
        for (int r = 0; r < 8; ++r) acc[nt][r] = bv;
    }
    #pragma unroll 1
    for (int kt = 0; kt < 16; ++kt) {
        const int k0 = kt * 4 + hi * 2;
        v2f A = *reinterpret_cast<const v2f*>(&S[m16 * RS + k0]);
        #pragma unroll
        for (int nt = 0; nt < 4; ++nt) {
            const int nb = nt * 16 + m16;
            v2f B;
            B.x = Wg[k0 * 64 + nb];
            B.y = Wg[(k0 + 1) * 64 + nb];
            acc[nt] = __builtin_amdgcn_wmma_f32_16x16x4_f32(
                false, A, false, B, (short)0, acc[nt], false, false);
        }
    }
    #pragma unroll
    for (int nt = 0; nt < 4; ++nt)
        #pragma unroll
        for (int r = 0; r < 8; ++r)
            S[(r + 8 * hi) * RS + (nt * 16 + m16)] = tanhf(acc[nt][r]);
}

__global__ __launch_bounds__(NW * 32) void kawahara_eq_kernel(
    const float* __restrict__ tx,
    const float* __restrict__ W0, const float* __restrict__ b0,
    const float* __restrict__ W1, const float* __restrict__ b1,
    const float* __restrict__ W2, const float* __restrict__ b2,
    const float* __restrict__ W3, const float* __restrict__ b3,
    const float* __restrict__ W4, const float* __restrict__ b4,
    float* __restrict__ out)
{
    // Wave-private activation jet planes S[c]: 16 points x 64 neurons, 6 coeffs
    __shared__ float S[NW][6][16 * RS];

    const int tid   = threadIdx.x;
    const int wv    = tid >> 5;
    const int lane  = tid & 31;
    const int m16   = lane & 15;
    const int hi    = lane >> 4;
    const int pbase = blockIdx.x * (NW * 16) + wv * 16;

    // ---------------- Layer 0: [2 -> 64], direct VALU + jet ----------------
    #pragma unroll 4
    for (int j = 0; j < 32; ++j) {
        int e = lane + 32 * j;          // 0..1023 over (m,n)
        int m = e >> 6;
        int n = e & 63;
        float t = tx[(pbase + m) * 2 + 0];
        float x = tx[(pbase + m) * 2 + 1];
        float a[6], y[6];
        a[0] = fmaf(t, W0[n], fmaf(x, W0[64 + n], b0[n]));
        a[1] = W0[64 + n];              // d(pre-act)/dx
        a[2] = 0.f; a[3] = 0.f; a[4] = 0.f; a[5] = 0.f;
        tanh_jet5(a, y);
        #pragma unroll
        for (int c = 0; c < 6; ++c) S[wv][c][m * RS + n] = y[c];
    }
    // S is wave-private; wave32 DS ops are in-order, compiler inserts dscnt waits.

    // ---------------- Layers 1..3: peeled so weights load as global ----------
    layer64_jet(W1, b1, S[wv], m16, hi);
    layer64_jet(W2, b2, S[wv], m16, hi);
    layer64_jet(W3, b3, S[wv], m16, hi);

    // ---------------- Layer 4: [64 -> 1] dot product ----------------
    float y[6] = { b4[0], 0.f, 0.f, 0.f, 0.f, 0.f };
    #pragma unroll 4
    for (int n = 0; n < 64; ++n) {
        float w = W4[n];
        #pragma unroll
        for (int c = 0; c < 6; ++c) y[c] = fmaf(S[wv][c][m16 * RS + n], w, y[c]);
    }

    if (hi == 0) {
        const int p      = pbase + m16;
        const float u    = y[0];
        const float ux   = y[1];
        const float ux3  = 6.0f   * y[3];     // 3! * Taylor coeff
        const float ux5  = 120.0f * y[5];     // 5! * Taylor coeff
        const float res  = 1.0f * ux3 + 0.25f * ux5
                         + (2.0f * u + 0.7500009999995948f) * ux;
        out[p]       = u;
        out[NEQ + p] = res;
    }
}

// Forward-only for init+boundary points.
__global__ __launch_bounds__(NW * 32) void kawahara_fwd_kernel(
    const float* __restrict__ tx_init, const float* __restrict__ tx_bound,
    const float* __restrict__ W0, const float* __restrict__ b0,
    const float* __restrict__ W1, const float* __restrict__ b1,
    const float* __restrict__ W2, const float* __restrict__ b2,
    const float* __restrict__ W3, const float* __restrict__ b3,
    const float* __restrict__ W4, const float* __restrict__ b4,
    float* __restrict__ out)
{
    __shared__ float S[NW][16 * RS];

    const int tid   = threadIdx.x;
    const int wv    = tid >> 5;
    const int lane  = tid & 31;
    const int m16   = lane & 15;
    const int hi    = lane >> 4;
    const int pbase = blockIdx.x * (NW * 16) + wv * 16;

    // Layer 0
    #pragma unroll 4
    for (int j = 0; j < 32; ++j) {
        int e = lane + 32 * j;
        int m = e >> 6;
        int n = e & 63;
        int p = pbase + m;
        const float* src = (p < 1024) ? (tx_init + p * 2) : (tx_bound + (p - 1024) * 2);
        float t = src[0], x = src[1];
        float a0 = fmaf(t, W0[n], fmaf(x, W0[64 + n], b0[n]));
        S[wv][m * RS + n] = tanhf(a0);
    }

    layer64_fwd(W1, b1, S[wv], m16, hi);
    layer64_fwd(W2, b2, S[wv], m16, hi);
    layer64_fwd(W3, b3, S[wv], m16, hi);

    float u = b4[0];
    #pragma unroll 4
    for (int n = 0; n < 64; ++n) u = fmaf(S[wv][m16 * RS + n], W4[n], u);

    if (hi == 0) out[2 * NEQ + pbase + m16] = u;
}

extern "C" void kernel_launch(void* const* d_in, const int* in_sizes, int n_in,
                              void* d_out, int out_size, void* d_ws, size_t ws_size,
                              hipStream_t stream) {
    const float* tx_eq    = (const float*)d_in[0];
    const float* tx_init  = (const float*)d_in[1];
    const float* tx_bound = (const float*)d_in[2];
    const float* W0 = (const float*)d_in[3];  const float* b0 = (const float*)d_in[4];
    const float* W1 = (const float*)d_in[5];  const float* b1 = (const float*)d_in[6];
    const float* W2 = (const float*)d_in[7];  const float* b2 = (const float*)d_in[8];
    const float* W3 = (const float*)d_in[9];  const float* b3 = (const float*)d_in[10];
    const float* W4 = (const float*)d_in[11]; const float* b4 = (const float*)d_in[12];
    float* out = (float*)d_out;

    // 32 points per block (2 waves x 16 points)
    kawahara_eq_kernel<<<NEQ / (NW * 16), NW * 32, 0, stream>>>(
        tx_eq, W0, b0, W1, b1, W2, b2, W3, b3, W4, b4, out);
    kawahara_fwd_kernel<<<NIB / (NW * 16), NW * 32, 0, stream>>>(
        tx_init, tx_bound, W0, b0, W1, b1, W2, b2, W3, b3, W4, b4, out);
}